// Dendrite_54417235641041
// MI455X (gfx1250) — compile-verified
//
#include <hip/hip_runtime.h>

typedef __attribute__((ext_vector_type(2))) float v2f;
typedef __attribute__((ext_vector_type(8))) float v8f;

// ds_swizzle group-of-32 XOR patterns: offset[14:10]=xor_mask, [4:0]=and_mask
template <int IMM>
__device__ __forceinline__ float swz_xor_add(float v) {
    int p = __builtin_amdgcn_ds_swizzle(__float_as_int(v), IMM);
    return v + __int_as_float(p);
}

// B=8192, D=32, S=8, M=255.  Rows = B*D = 262144, 16 rows per wave-tile.
__global__ __launch_bounds__(256) void dendrite_wmma_kernel(
    const float* __restrict__ x,      // [B, S]
    const float* __restrict__ kk,     // [D, S]
    const float* __restrict__ w,      // [D, S]
    const float* __restrict__ q,      // [D, S]
    const float* __restrict__ lin_w,  // [255]
    const float* __restrict__ lin_b,  // [1]
    float* __restrict__ out)          // [B*D]
{
    __shared__ float plo_lds[8][16][16];   // [wave][row][i] : 8 KB / block

    const int lane = threadIdx.x & 31;
    const int wv   = threadIdx.x >> 5;
    const int half = lane >> 4;            // 0: lanes 0-15, 1: lanes 16-31
    const int n    = lane & 15;
    const int tile = blockIdx.x * 8 + wv;
    const int row_g0 = tile * 16;

    // ---- B operand: W2[h][i] = lin_w[16h+i-1] (0 when h=i=0) -----------
    // B 4x16 f32 layout: lane holds col n=l%16; vgpr e holds K=4j+2*half+e
    v2f bmat[4];
    #pragma unroll
    for (int j = 0; j < 4; ++j) {
        #pragma unroll
        for (int e = 0; e < 2; ++e) {
            int h = 4 * j + 2 * half + e;
            int v = h * 16 + n;
            bmat[j][e] = (v == 0) ? 0.0f : lin_w[v - 1];
        }
    }

    // ---- 8 sigmoids for this lane's row (two lanes share a row) --------
    const int row_g = row_g0 + n;
    const int b_idx = row_g >> 5;          // D == 32
    const int d_idx = row_g & 31;

    float sv[8];
    #pragma unroll
    for (int s = 0; s < 8; ++s) {
        float xs = x[b_idx * 8 + s];
        float z  = kk[d_idx * 8 + s] * (w[d_idx * 8 + s] * xs - q[d_idx * 8 + s]);
        sv[s] = __builtin_amdgcn_rcpf(1.0f + __expf(-z));  // sigmoid(z)
    }

    // ---- phi subset products, only the 8 values this lane feeds to A --
    // ph[h] = prod(t_j : bit j of h), t0=sv[3], t1=sv[2], t2=sv[1], t3=sv[0]
    // lane needs h = 2*half + {0,1,4,5,8,9,12,13} -> base * {1,t0,t2,...}
    const float t0 = sv[3], t1 = sv[2], t2 = sv[1], t3 = sv[0];
    const float base = half ? t1 : 1.0f;   // single cndmask

    v2f amat[4];
    amat[0][0] = base;                     // h = c
    amat[0][1] = base * t0;                // h = c+1
    amat[1][0] = base * t2;                // h = c+4
    amat[1][1] = amat[1][0] * t0;          // h = c+5
    amat[2][0] = base * t3;                // h = c+8
    amat[2][1] = amat[2][0] * t0;          // h = c+9
    amat[3][0] = amat[2][0] * t2;          // h = c+12
    amat[3][1] = amat[3][0] * t0;          // h = c+13

    // ---- plo subset products (s = 7..4), staged via LDS ----------------
    if (half == 0) {
        float pl[16];
        pl[0] = 1.0f;
        int len = 1;
        #pragma unroll
        for (int j = 0; j < 4; ++j) {
            float tl = sv[7 - j];
            #pragma unroll
            for (int m = 0; m < 8; ++m) {
                if (m < len) pl[len + m] = pl[m] * tl;   // folds at compile time
            }
            len <<= 1;
        }
        #pragma unroll
        for (int i = 0; i < 16; ++i)
            plo_lds[wv][n][i] = pl[i];
    }

    // ---- Y = Phi @ W2 via 4 chained V_WMMA_F32_16X16X4_F32 -------------
    v8f c = {};
    #pragma unroll
    for (int j = 0; j < 4; ++j)
        c = __builtin_amdgcn_wmma_f32_16x16x4_f32(
                /*neg_a=*/false, amat[j], /*neg_b=*/false, bmat[j],
                /*c_mod=*/(short)0, c, /*reuse_a=*/false, /*reuse_b=*/false);

    // ---- out[row] = sum_i Y[row,i] * plo[row,i] + bias -----------------
    // D layout: vgpr r, this lane holds Y[r + 8*half][n]
    float partial[8];
    #pragma unroll
    for (int r = 0; r < 8; ++r)
        partial[r] = c[r] * plo_lds[wv][8 * half + r][n];

    // xor-reduce over the 16 lanes of each half (SWAPX1/2/4/8 swizzles)
    #pragma unroll
    for (int r = 0; r < 8; ++r) {
        float v = partial[r];
        v = swz_xor_add<(1 << 10) | 0x1f>(v);
        v = swz_xor_add<(2 << 10) | 0x1f>(v);
        v = swz_xor_add<(4 << 10) | 0x1f>(v);
        v = swz_xor_add<(8 << 10) | 0x1f>(v);
        partial[r] = v;
    }

    if (n == 0) {                           // lanes 0 and 16 write 8 rows each
        float bias = lin_b[0];
        #pragma unroll
        for (int r = 0; r < 8; ++r)
            out[row_g0 + 8 * half + r] = partial[r] + bias;
    }
}

extern "C" void kernel_launch(void* const* d_in, const int* in_sizes, int n_in,
                              void* d_out, int out_size, void* d_ws, size_t ws_size,
                              hipStream_t stream) {
    // inputs: x, k, w, q, mask(unused - regenerated from bit math), lin_w, lin_b
    const float* x     = (const float*)d_in[0];
    const float* k     = (const float*)d_in[1];
    const float* w     = (const float*)d_in[2];
    const float* q     = (const float*)d_in[3];
    const float* lin_w = (const float*)d_in[5];
    const float* lin_b = (const float*)d_in[6];
    float* out = (float*)d_out;

    // 262144 rows / 16 rows-per-wave / 8 waves-per-block = 2048 blocks
    dendrite_wmma_kernel<<<dim3(2048), dim3(256), 0, stream>>>(
        x, k, w, q, lin_w, lin_b, out);
}